// TrueOctonionMultiply_43224550867321
// MI455X (gfx1250) — compile-verified
//
#include <hip/hip_runtime.h>

// ---------------------------------------------------------------------------
// TrueOctonionMultiply — memory-bound streaming kernel for MI455X (gfx1250).
//
// Roofline: B = 4,194,304 octonion pairs, 96 B HBM traffic and ~120 f32 FLOPs
// per pair -> 402 MB total @ 23.3 TB/s = ~17.3 us floor; compute is well under
// that. So: pure HBM stream. b128 loads/stores with non-temporal hints (stream
// exceeds the 192 MB L2), 2 octonions per thread with all 8 b128 loads issued
// up front (~4 KB in flight per wave -> ~64 KB per SIMD at high occupancy,
// covering HBM latency at the per-SIMD bandwidth share). wave32, 256-thread
// blocks (8 waves). WMMA is deliberately not used: the per-sample sparse
// 8x8x8 structure-constant contraction would require materializing an 8x8
// matrix per element — more VALU and register traffic, zero bytes saved.
// ---------------------------------------------------------------------------

typedef float v4f __attribute__((ext_vector_type(4)));

// Octonion product per the reference's Fano triples
// _FANO = [(0,1,2),(0,3,4),(0,6,5),(1,3,5),(1,4,6),(2,3,6),(2,5,4)]
__device__ __forceinline__ void octo_mul(const v4f al, const v4f ah,
                                         const v4f bl, const v4f bh,
                                         v4f& ol, v4f& oh) {
  const float a0 = al.x, a1 = al.y, a2 = al.z, a3 = al.w;
  const float a4 = ah.x, a5 = ah.y, a6 = ah.z, a7 = ah.w;
  const float b0 = bl.x, b1 = bl.y, b2 = bl.z, b3 = bl.w;
  const float b4 = bh.x, b5 = bh.y, b6 = bh.z, b7 = bh.w;

  ol.x = a0*b0 - a1*b1 - a2*b2 - a3*b3 - a4*b4 - a5*b5 - a6*b6 - a7*b7;
  ol.y = a0*b1 + a1*b0 + a2*b3 - a3*b2 + a4*b5 - a5*b4 + a7*b6 - a6*b7;
  ol.z = a0*b2 + a2*b0 + a3*b1 - a1*b3 + a4*b6 - a6*b4 + a5*b7 - a7*b5;
  ol.w = a0*b3 + a3*b0 + a1*b2 - a2*b1 + a4*b7 - a7*b4 + a6*b5 - a5*b6;
  oh.x = a0*b4 + a4*b0 + a5*b1 - a1*b5 + a6*b2 - a2*b6 + a7*b3 - a3*b7;
  oh.y = a0*b5 + a5*b0 + a1*b4 - a4*b1 + a3*b6 - a6*b3 + a7*b2 - a2*b7;
  oh.z = a0*b6 + a6*b0 + a1*b7 - a7*b1 + a2*b4 - a4*b2 + a5*b3 - a3*b5;
  oh.w = a0*b7 + a7*b0 + a6*b1 - a1*b6 + a2*b5 - a5*b2 + a3*b4 - a4*b3;
}

// Hot path: n is an exact multiple of blockDim*2 (true for B = 2^22).
// No guards, no exec manipulation, no VGPR zero-init: straight-line
// 8x global_load_b128 NT -> 2x octonion product -> 4x global_store_b128 NT.
__global__ __launch_bounds__(256)
void octo_mul_kernel_exact(const v4f* __restrict__ a, const v4f* __restrict__ b,
                           v4f* __restrict__ out) {
  const size_t T  = (size_t)gridDim.x * blockDim.x;
  const size_t i0 = (size_t)blockIdx.x * blockDim.x + threadIdx.x;
  const size_t i1 = i0 + T;

  const v4f* pa0 = a + 2 * i0;
  const v4f* pb0 = b + 2 * i0;
  const v4f* pa1 = a + 2 * i1;
  const v4f* pb1 = b + 2 * i1;

  const v4f al0 = __builtin_nontemporal_load(pa0);
  const v4f ah0 = __builtin_nontemporal_load(pa0 + 1);
  const v4f bl0 = __builtin_nontemporal_load(pb0);
  const v4f bh0 = __builtin_nontemporal_load(pb0 + 1);
  const v4f al1 = __builtin_nontemporal_load(pa1);
  const v4f ah1 = __builtin_nontemporal_load(pa1 + 1);
  const v4f bl1 = __builtin_nontemporal_load(pb1);
  const v4f bh1 = __builtin_nontemporal_load(pb1 + 1);

  v4f ol0, oh0, ol1, oh1;
  octo_mul(al0, ah0, bl0, bh0, ol0, oh0);
  octo_mul(al1, ah1, bl1, bh1, ol1, oh1);

  v4f* po0 = out + 2 * i0;
  v4f* po1 = out + 2 * i1;
  __builtin_nontemporal_store(ol0, po0);
  __builtin_nontemporal_store(oh0, po0 + 1);
  __builtin_nontemporal_store(ol1, po1);
  __builtin_nontemporal_store(oh1, po1 + 1);
}

// Tail-safe fallback for n not divisible by (blockDim * 2).
__global__ __launch_bounds__(256)
void octo_mul_kernel_guarded(const v4f* __restrict__ a, const v4f* __restrict__ b,
                             v4f* __restrict__ out, int n) {
  const int tid = blockIdx.x * blockDim.x + threadIdx.x;
  const int T   = gridDim.x * blockDim.x;
  const int i0  = tid;
  const int i1  = tid + T;

  if (i0 < n) {
    const v4f* pa = a + 2 * (size_t)i0;
    const v4f* pb = b + 2 * (size_t)i0;
    const v4f al = __builtin_nontemporal_load(pa);
    const v4f ah = __builtin_nontemporal_load(pa + 1);
    const v4f bl = __builtin_nontemporal_load(pb);
    const v4f bh = __builtin_nontemporal_load(pb + 1);
    v4f ol, oh;
    octo_mul(al, ah, bl, bh, ol, oh);
    v4f* po = out + 2 * (size_t)i0;
    __builtin_nontemporal_store(ol, po);
    __builtin_nontemporal_store(oh, po + 1);
  }
  if (i1 < n) {
    const v4f* pa = a + 2 * (size_t)i1;
    const v4f* pb = b + 2 * (size_t)i1;
    const v4f al = __builtin_nontemporal_load(pa);
    const v4f ah = __builtin_nontemporal_load(pa + 1);
    const v4f bl = __builtin_nontemporal_load(pb);
    const v4f bh = __builtin_nontemporal_load(pb + 1);
    v4f ol, oh;
    octo_mul(al, ah, bl, bh, ol, oh);
    v4f* po = out + 2 * (size_t)i1;
    __builtin_nontemporal_store(ol, po);
    __builtin_nontemporal_store(oh, po + 1);
  }
}

extern "C" void kernel_launch(void* const* d_in, const int* in_sizes, int n_in,
                              void* d_out, int out_size, void* d_ws, size_t ws_size,
                              hipStream_t stream) {
  (void)n_in; (void)out_size; (void)d_ws; (void)ws_size;

  const v4f* a = (const v4f*)d_in[0];
  const v4f* b = (const v4f*)d_in[1];
  v4f* out     = (v4f*)d_out;

  const int n     = in_sizes[0] / 8;   // number of octonions (B = 4,194,304)
  const int BLOCK = 256;               // 8 wave32 waves per block
  const int EPT   = 2;                 // octonions per thread

  if (n % (BLOCK * EPT) == 0) {
    const int blocks = n / (BLOCK * EPT);   // 8192 blocks for B = 2^22
    octo_mul_kernel_exact<<<blocks, BLOCK, 0, stream>>>(a, b, out);
  } else {
    const int blocks = (n + BLOCK * EPT - 1) / (BLOCK * EPT);
    octo_mul_kernel_guarded<<<blocks, BLOCK, 0, stream>>>(a, b, out, n);
  }
}